// ChamferLoss_34815004901800
// MI455X (gfx1250) — compile-verified
//
#include <hip/hip_runtime.h>
#include <math.h>

// Chamfer loss, B=16, N=M=2048, D=4.
// dist^2[i,j] = ||x_i||^2 + ||y_j||^2 - 2 * x_i . y_j   (eps=1e-12 << fp32 ulp)
// Computed ENTIRELY inside V_WMMA_F32_16X16X4_F32 (K=4 == D):
//   A' = -2x  (16x4),  B = y^T (4x16),  C[v] = ||x_v||^2 + ||y_j||^2
//   => D = A'xB + C = d^2 tile, straight out of the matrix pipe.
// Mins tracked on squared distances (sqrt is monotone, commutes with min);
// sqrt applied once per final min in the reduce kernel.

typedef float v2f __attribute__((ext_vector_type(2)));
typedef float v8f __attribute__((ext_vector_type(8)));

#define CB  16                   // batches
#define CN  2048                 // x points per batch
#define CM  2048                 // y points per batch
#define CD  4                    // dims
#define JT  (CM / 16)            // 128 column tiles
#define TROWS (CB * (CN / 16))   // 2048 tile-rows
#define INF_BITS 0x7F800000u

// ---------------- init: +inf into min arrays, zero the scalar output -------
__global__ void chamfer_init_kernel(unsigned int* mins, int total, float* out) {
    int idx = blockIdx.x * blockDim.x + threadIdx.x;
    if (idx == 0) out[0] = 0.0f;
    if (idx < total) mins[idx] = INF_BITS;   // +inf as f32 bits
}

// ---------------- main: WMMA pairwise d^2 + running mins -------------------
__global__ void __launch_bounds__(256)
chamfer_wmma_kernel(const float* __restrict__ x, const float* __restrict__ y,
                    float* __restrict__ min_xy,            // [CB*CN] d^2
                    unsigned int* __restrict__ min_yx) {   // [CB*CM] d^2 bits
    __shared__ float        syn[CM];   // ||y_j||^2, computed once per block
    __shared__ unsigned int scol[CM];  // block-local column min (d^2 bits)

    const int b = blockIdx.x >> 4;     // 16 blocks per batch (128 tile-rows / 8)
    const float* yb = y + (size_t)b * CM * CD;

    for (int j = threadIdx.x; j < CM; j += 256) {
        const float4 yv = *(const float4*)(yb + (size_t)j * CD);
        syn[j]  = yv.x * yv.x + yv.y * yv.y + yv.z * yv.z + yv.w * yv.w;
        scol[j] = INF_BITS;
    }
    __syncthreads();

    const int wave = threadIdx.x >> 5;
    const int lane = threadIdx.x & 31;
    const int tileRow = blockIdx.x * 8 + wave;   // 8 tile-rows per block
    const int I  = tileRow & 127;                // tile-row within batch
    const int hi = lane >> 4;                    // 0: lanes 0-15, 1: lanes 16-31
    const int lo = lane & 15;

    const float* xb = x + (size_t)b * CN * CD;

    // A fragment, pre-scaled by -2: lane holds -2*x[I*16 + lo, 2*hi + {0,1}]
    const int arow = I * 16 + lo;
    v2f afrag;
    afrag.x = -2.0f * xb[arow * CD + 2 * hi + 0];
    afrag.y = -2.0f * xb[arow * CD + 2 * hi + 1];

    // ||x||^2 for the 8 C-layout rows this lane covers: r = v + 8*hi
    float xn[8];
#pragma unroll
    for (int v = 0; v < 8; ++v) {
        const float4 xv = *(const float4*)(xb + (size_t)(I * 16 + v + 8 * hi) * CD);
        xn[v] = xv.x * xv.x + xv.y * xv.y + xv.z * xv.z + xv.w * xv.w;
    }

    float rowMin[8];
#pragma unroll
    for (int v = 0; v < 8; ++v) rowMin[v] = INFINITY;

    for (int J = 0; J < JT; ++J) {
        const int jrow = J * 16 + lo;
        // B fragment loaded directly: y[jrow, 2*hi + {0,1}] (8B aligned)
        const v2f bfrag = *(const v2f*)(yb + (size_t)jrow * CD + 2 * hi);
        const float yn  = syn[jrow];

        // Seed the accumulator: C[v] = ||x_v||^2 + ||y_j||^2
        v8f cacc;
#pragma unroll
        for (int v = 0; v < 8; ++v) cacc[v] = xn[v] + yn;

        // D = (-2x) . y + (xn + yn) = d^2, computed by the matrix pipe
        const v8f d2 = __builtin_amdgcn_wmma_f32_16x16x4_f32(
                /*neg_a=*/false, afrag, /*neg_b=*/false, bfrag,
                /*c_mod=*/(short)0, cacc, /*reuse_a=*/false, /*reuse_b=*/false);

        float colMin = INFINITY;
#pragma unroll
        for (int v = 0; v < 8; ++v) {
            rowMin[v] = fminf(rowMin[v], d2[v]);
            colMin    = fminf(colMin, d2[v]);
        }
        // clamp once (max(.,0) monotone, commutes with min); both halves
        // min into the same LDS slot -> no cross-half shuffle needed
        colMin = fmaxf(colMin, 0.0f);
        atomicMin(&scol[jrow], __float_as_uint(colMin));
    }

    // row mins: reduce across the 16 lanes (columns) within each half
#pragma unroll
    for (int m = 8; m >= 1; m >>= 1) {
#pragma unroll
        for (int v = 0; v < 8; ++v)
            rowMin[v] = fminf(rowMin[v], __shfl_xor(rowMin[v], m, 32));
    }
    if (lo == 0) {
#pragma unroll
        for (int v = 0; v < 8; ++v)
            min_xy[(size_t)b * CN + I * 16 + v + 8 * hi] = fmaxf(rowMin[v], 0.0f);
    }

    // commit block-local column mins: one global atomic per column per block
    __syncthreads();
    for (int j = threadIdx.x; j < CM; j += 256)
        atomicMin(&min_yx[(size_t)b * CM + j], scol[j]);
}

// ---------------- final: sum sqrt(d^2) over min_xy and min_yx --------------
__global__ void chamfer_reduce_kernel(const float* __restrict__ vals, int n,
                                      float* __restrict__ out) {
    __shared__ float sdata[256];
    float s = 0.0f;
    for (int i = blockIdx.x * blockDim.x + threadIdx.x; i < n;
         i += gridDim.x * blockDim.x)
        s += __builtin_amdgcn_sqrtf(vals[i]);    // raw v_sqrt_f32, ~1 ulp
    sdata[threadIdx.x] = s;
    __syncthreads();
    for (int stride = 128; stride > 0; stride >>= 1) {
        if (threadIdx.x < stride) sdata[threadIdx.x] += sdata[threadIdx.x + stride];
        __syncthreads();
    }
    if (threadIdx.x == 0) atomicAdd(out, sdata[0]);
}

extern "C" void kernel_launch(void* const* d_in, const int* in_sizes, int n_in,
                              void* d_out, int out_size, void* d_ws, size_t ws_size,
                              hipStream_t stream) {
    const float* x = (const float*)d_in[0];   // [B, N, 4]
    const float* y = (const float*)d_in[1];   // [B, M, 4]
    float* out = (float*)d_out;               // scalar
    float* mins_f = (float*)d_ws;             // [B*N] min_xy | [B*M] min_yx (256 KB)
    unsigned int* mins_u = (unsigned int*)d_ws;
    float*        min_xy = mins_f;
    unsigned int* min_yx = mins_u + (size_t)CB * CN;
    const int total = CB * CN + CB * CM;

    chamfer_init_kernel<<<(total + 255) / 256, 256, 0, stream>>>(mins_u, total, out);
    chamfer_wmma_kernel<<<TROWS / 8, 256, 0, stream>>>(x, y, min_xy, min_yx);
    chamfer_reduce_kernel<<<64, 256, 0, stream>>>(mins_f, total, out);
}